// MutualModule1_28819230556464
// MI455X (gfx1250) — compile-verified
//
#include <hip/hip_runtime.h>
#include <cstdint>

#define DIM  64
#define NREG 128
#define VB   (64*64*64)   // 262144 voxels
#define EPSV 1e-5f

typedef __attribute__((ext_vector_type(2))) float v2f;
typedef __attribute__((ext_vector_type(8))) float v8f;

static __device__ __forceinline__ v8f wmma4(v2f a, v2f b, v8f c) {
  // V_WMMA_F32_16X16X4_F32 : D = A(16x4) x B(4x16) + C(16x16), all f32
  return __builtin_amdgcn_wmma_f32_16x16x4_f32(false, a, false, b, (short)0, c,
                                               false, false);
}

// ---- workspace layout (float offsets) ----
// Weight operands stored PRE-SWIZZLED in WMMA A-fragment order:
//   frag(s, mt, lane) = { W[m][4s+2h+0], W[m][4s+2h+1] }  with m = mt*16+(lane&15), h = lane>>4
// so kernel 2 fetches one coalesced b64 per fragment with constant ioffset.
constexpr int OFF_W0A = 0;                       // [2][32*4][32][2]  = 16384
constexpr int OFF_W1A = OFF_W0A + 2*NREG*DIM;    // [16*4][32][2]     = 4096
constexpr int OFF_P0A = OFF_W1A + DIM*DIM;       // [16][32][2]       = 1024
constexpr int OFF_P1A = OFF_P0A + DIM*16;        // [16][32][2]       = 1024
constexpr int OFF_SH0 = OFF_P1A + DIM*16;        // [64] bn0 shift
constexpr int OFF_SH1 = OFF_SH0 + DIM;           // [64] bn1 shift
constexpr int OFF_X   = OFF_SH1 + DIM;           // [2][64][128] GCN ping
constexpr int OFF_Y   = OFF_X + 2*DIM*NREG;      // [2][64][128] GCN pong
constexpr int OFF_SUP = OFF_Y + 2*DIM*NREG;      // [2][128][64] support
constexpr int OFF_S   = OFF_SUP + 2*NREG*DIM;    // [2][128][128] adj scores
// total = 104576 floats ~= 409 KB

// ---------------------------------------------------------------------------
// Kernel 1: GCN (loop=3) + fold BN/conv into streaming operands, emitted in
// WMMA-A fragment-interleaved order. Tiny (one block per batch).
// ---------------------------------------------------------------------------
__global__ __launch_bounds__(256) void gcn_prep_kernel(
    const float* __restrict__ rg,
    const float* __restrict__ w1g, const float* __restrict__ w2g,
    const float* __restrict__ w3g,
    const float* __restrict__ pred0_w, const float* __restrict__ pred1_w,
    const float* __restrict__ conv0_w,
    const float* __restrict__ g0, const float* __restrict__ b0,
    const float* __restrict__ m0, const float* __restrict__ v0,
    const float* __restrict__ conv1_w,
    const float* __restrict__ g1, const float* __restrict__ b1,
    const float* __restrict__ m1, const float* __restrict__ v1,
    float* __restrict__ ws)
{
  const int b = blockIdx.x;
  const int t = threadIdx.x;

  float* X   = ws + OFF_X   + b*DIM*NREG;   // [c][n]
  float* Y   = ws + OFF_Y   + b*DIM*NREG;
  float* SUP = ws + OFF_SUP + b*NREG*DIM;   // [n][d]
  float* S   = ws + OFF_S   + b*NREG*NREG;  // [n][m]

  const float* xin = rg + b*DIM*NREG;
  for (int i = t; i < DIM*NREG; i += 256) X[i] = xin[i];
  __syncthreads();

  const float* wlist[3] = {w1g, w2g, w3g};
  for (int it = 0; it < 3; ++it) {
    const float* w = wlist[it];
    // scores S[n][m] = sum_c X[c][n] * X[c][m]
    for (int e = t; e < NREG*NREG; e += 256) {
      int n = e >> 7, m = e & 127;
      float s = 0.f;
      for (int c = 0; c < DIM; ++c) s += X[c*NREG + n] * X[c*NREG + m];
      S[e] = s;
    }
    __syncthreads();
    // row softmax (axis m)
    if (t < NREG) {
      float* row = S + t*NREG;
      float mx = row[0];
      for (int m = 1; m < NREG; ++m) mx = fmaxf(mx, row[m]);
      float sum = 0.f;
      for (int m = 0; m < NREG; ++m) { float e_ = __expf(row[m]-mx); row[m] = e_; sum += e_; }
      float inv = 1.f/sum;
      for (int m = 0; m < NREG; ++m) row[m] *= inv;
    }
    // SUP[n][d] = sum_c X[c][n] * w[c][d]
    for (int e = t; e < NREG*DIM; e += 256) {
      int n = e >> 6, d = e & 63;
      float s = 0.f;
      for (int c = 0; c < DIM; ++c) s += X[c*NREG + n] * w[c*DIM + d];
      SUP[e] = s;
    }
    __syncthreads();
    // Y[d][n] = sum_m S[n][m] * SUP[m][d]
    for (int e = t; e < DIM*NREG; e += 256) {
      int d = e >> 7, n = e & 127;
      float s = 0.f;
      const float* srow = S + n*NREG;
      for (int m = 0; m < NREG; ++m) s += srow[m] * SUP[m*DIM + d];
      Y[d*NREG + n] = s;
    }
    __syncthreads();
    float* tmp = X; X = Y; Y = tmp;
  }

  // W0[m][k] = inv0[m] * sum_c conv0_w[m][c]*relu(G[c][k]) -> A-frag order
  float* W0A = ws + OFF_W0A + b*NREG*DIM;
  for (int e = t; e < NREG*DIM; e += 256) {
    int k = e >> 6, m = e & 63;
    float inv = g0[m] * rsqrtf(v0[m] + EPSV);
    float s = 0.f;
    for (int c = 0; c < DIM; ++c) s += conv0_w[m*DIM + c] * fmaxf(X[c*NREG + k], 0.f);
    int sidx = k >> 2, kr = k & 3;
    int lane = (m & 15) | ((kr >> 1) << 4);
    int mt = m >> 4, r = kr & 1;
    W0A[(((sidx*4 + mt)*32) + lane)*2 + r] = inv * s;
  }

  if (b == 0) {
    float* W1A = ws + OFF_W1A;
    float* P0A = ws + OFF_P0A;
    float* P1A = ws + OFF_P1A;
    float* SH0 = ws + OFF_SH0;
    float* SH1 = ws + OFF_SH1;
    for (int e = t; e < DIM*DIM; e += 256) {
      int c = e >> 6, m = e & 63;
      float inv = g1[m] * rsqrtf(v1[m] + EPSV);
      int sidx = c >> 2, kr = c & 3;
      int lane = (m & 15) | ((kr >> 1) << 4);
      int mt = m >> 4, r = kr & 1;
      W1A[(((sidx*4 + mt)*32) + lane)*2 + r] = inv * conv1_w[m*DIM + c];
    }
    for (int e = t; e < DIM*16; e += 256) {
      int c = e >> 4, m = e & 15;
      int sidx = c >> 2, kr = c & 3;
      int lane = m | ((kr >> 1) << 4);
      int r = kr & 1;
      int idx = ((sidx*32) + lane)*2 + r;
      P0A[idx] = (m < 3) ? pred0_w[m*DIM + c] : 0.f;
      P1A[idx] = (m < 3) ? pred1_w[m*DIM + c] : 0.f;
    }
    if (t < DIM) {
      float i0 = g0[t] * rsqrtf(v0[t] + EPSV);
      float i1 = g1[t] * rsqrtf(v1[t] + EPSV);
      SH0[t] = b0[t] - m0[t]*i0;
      SH1[t] = b1[t] - m1[t]*i1;
    }
  }
}

// ---------------------------------------------------------------------------
// Kernel 2: fully fused streaming pass. One wave = one 16-voxel tile.
// All weight A-fragments arrive as single coalesced b64 loads with constant
// ioffsets; LDS round-trips use one base VGPR + immediate ds offsets.
// ---------------------------------------------------------------------------
__global__ __launch_bounds__(256) void fused_main_kernel(
    const float* __restrict__ region_x,   // [B][64][V]
    const float* __restrict__ assign,     // [B][128][V]
    const float* __restrict__ contour_x,  // [B][64][V]
    const float* __restrict__ pred0_b,    // [3]
    const float* __restrict__ pred1_b,    // [3]
    const float* __restrict__ ws,
    float* __restrict__ out)              // [contour 2*3*V | region 2*3*V]
{
  __shared__ float lds[8 * DIM * 16];   // 32 KB, 4KB per wave (wave-private)

  const int tilesPerBatch = VB / 128;
  const int b        = blockIdx.x / tilesPerBatch;
  const int voxblock = (blockIdx.x % tilesPerBatch) * 128;
  const int wave = threadIdx.x >> 5;
  const int lane = threadIdx.x & 31;
  const int n = lane & 15;          // voxel column within tile / A-row index
  const int h = lane >> 4;          // lane half
  const size_t vox = (size_t)voxblock + wave*16 + n;

  float* myLds = lds + wave * DIM * 16;        // [64][16]
  float*       stB = myLds + h*128 + n;        // C-layout store base
  const float* ldB = myLds + h*32  + n;        // B-frag load base

  const v2f* W0A = (const v2f*)(ws + OFF_W0A) + (size_t)b*(NREG*DIM/2) + lane;
  const v2f* W1A = (const v2f*)(ws + OFF_W1A) + lane;
  const v2f* P0A = (const v2f*)(ws + OFF_P0A) + lane;
  const v2f* P1A = (const v2f*)(ws + OFF_P1A) + lane;
  const float* SH0 = ws + OFF_SH0 + 8*h;
  const float* SH1 = ws + OFF_SH1 + 8*h;

  const float* cxb = contour_x + (size_t)b*DIM*VB + (size_t)8*h*VB + vox;
  const float* rxb = region_x  + (size_t)b*DIM*VB + (size_t)8*h*VB + vox;
  const float* abh = assign    + (size_t)b*NREG*VB + (size_t)2*h*VB + vox;

  // ---- load contour_x tile in C-layout regs and stash into LDS ----
  v8f cx[4];
  #pragma unroll
  for (int mt = 0; mt < 4; ++mt) {
    const float* cb = cxb + (size_t)mt*16*VB;
    #pragma unroll
    for (int g = 0; g < 8; ++g) {
      float v = cb[(size_t)g*VB];         // ioffset-folded strided load
      cx[mt][g] = v;
      stB[(mt*16 + g)*16] = v;            // immediate-offset ds_store
    }
  }

  // ---- contour = pred0 @ contour_x + b0 ----
  v8f accp;
  #pragma unroll
  for (int g = 0; g < 8; ++g) accp[g] = 0.f;
  #pragma unroll
  for (int s = 0; s < 16; ++s) {
    v2f bf; bf.x = ldB[(4*s+0)*16]; bf.y = ldB[(4*s+1)*16];
    accp = wmma4(P0A[s*32], bf, accp);
  }
  if (h == 0) {
    #pragma unroll
    for (int o = 0; o < 3; ++o)
      out[((size_t)b*3 + o)*VB + vox] = accp[o] + pred0_b[o];
  }

  // ---- T = W0 @ assign_cols   (64x128 * 128x16) ----
  v8f acc[4];
  #pragma unroll
  for (int mt = 0; mt < 4; ++mt)
    #pragma unroll
    for (int g = 0; g < 8; ++g) acc[mt][g] = 0.f;

  #pragma unroll
  for (int s = 0; s < 32; ++s) {
    const float* as = abh + (size_t)4*s*VB;
    v2f bf; bf.x = as[0]; bf.y = as[VB];
    #pragma unroll
    for (int mt = 0; mt < 4; ++mt)
      acc[mt] = wmma4(W0A[(s*4 + mt)*32], bf, acc[mt]);
  }

  // ---- rx = relu(T + sh0) + contour_x + region_x ; stash to LDS ----
  #pragma unroll
  for (int mt = 0; mt < 4; ++mt) {
    const float* rb = rxb + (size_t)mt*16*VB;
    #pragma unroll
    for (int g = 0; g < 8; ++g) {
      float t = fmaxf(acc[mt][g] + SH0[mt*16 + g], 0.f);
      t += cx[mt][g] + rb[(size_t)g*VB];
      stB[(mt*16 + g)*16] = t;
    }
  }

  // ---- u = relu(W1 @ rx + sh1) ----
  v8f acc2[4];
  #pragma unroll
  for (int mt = 0; mt < 4; ++mt)
    #pragma unroll
    for (int g = 0; g < 8; ++g) acc2[mt][g] = 0.f;

  #pragma unroll
  for (int s = 0; s < 16; ++s) {
    v2f bf; bf.x = ldB[(4*s+0)*16]; bf.y = ldB[(4*s+1)*16];
    #pragma unroll
    for (int mt = 0; mt < 4; ++mt)
      acc2[mt] = wmma4(W1A[(s*4 + mt)*32], bf, acc2[mt]);
  }
  #pragma unroll
  for (int mt = 0; mt < 4; ++mt)
    #pragma unroll
    for (int g = 0; g < 8; ++g)
      stB[(mt*16 + g)*16] = fmaxf(acc2[mt][g] + SH1[mt*16 + g], 0.f);

  // ---- region = pred1 @ u + b1 ----
  v8f accq;
  #pragma unroll
  for (int g = 0; g < 8; ++g) accq[g] = 0.f;
  #pragma unroll
  for (int s = 0; s < 16; ++s) {
    v2f bf; bf.x = ldB[(4*s+0)*16]; bf.y = ldB[(4*s+1)*16];
    accq = wmma4(P1A[s*32], bf, accq);
  }
  if (h == 0) {
    float* outr = out + (size_t)2*3*VB;   // region section
    #pragma unroll
    for (int o = 0; o < 3; ++o)
      outr[((size_t)b*3 + o)*VB + vox] = accq[o] + pred1_b[o];
  }
}

// ---------------------------------------------------------------------------
extern "C" void kernel_launch(void* const* d_in, const int* in_sizes, int n_in,
                              void* d_out, int out_size, void* d_ws, size_t ws_size,
                              hipStream_t stream) {
  const float* region_x     = (const float*)d_in[0];
  const float* region_graph = (const float*)d_in[1];
  const float* assign       = (const float*)d_in[2];
  const float* contour_x    = (const float*)d_in[3];
  const float* gcn_w1       = (const float*)d_in[4];
  const float* gcn_w2       = (const float*)d_in[5];
  const float* gcn_w3       = (const float*)d_in[6];
  const float* pred0_w      = (const float*)d_in[7];
  const float* pred0_b      = (const float*)d_in[8];
  const float* pred1_w      = (const float*)d_in[9];
  const float* pred1_b      = (const float*)d_in[10];
  const float* conv0_w      = (const float*)d_in[11];
  const float* bn0_gamma    = (const float*)d_in[12];
  const float* bn0_beta     = (const float*)d_in[13];
  const float* bn0_mean     = (const float*)d_in[14];
  const float* bn0_var      = (const float*)d_in[15];
  const float* conv1_w      = (const float*)d_in[16];
  const float* bn1_gamma    = (const float*)d_in[17];
  const float* bn1_beta     = (const float*)d_in[18];
  const float* bn1_mean     = (const float*)d_in[19];
  const float* bn1_var      = (const float*)d_in[20];

  float* ws  = (float*)d_ws;
  float* out = (float*)d_out;

  gcn_prep_kernel<<<2, 256, 0, stream>>>(
      region_graph, gcn_w1, gcn_w2, gcn_w3, pred0_w, pred1_w,
      conv0_w, bn0_gamma, bn0_beta, bn0_mean, bn0_var,
      conv1_w, bn1_gamma, bn1_beta, bn1_mean, bn1_var, ws);

  fused_main_kernel<<<2 * (VB/128), 256, 0, stream>>>(
      region_x, assign, contour_x, pred0_b, pred1_b, ws, out);
}